// GIN_23828478558294
// MI455X (gfx1250) — compile-verified
//
#include <hip/hip_runtime.h>
#include <stdint.h>

#define DIM 128

typedef __attribute__((ext_vector_type(2))) float v2f;
typedef __attribute__((ext_vector_type(4))) float f32x4;
typedef __attribute__((ext_vector_type(8))) float v8f;
typedef __attribute__((ext_vector_type(4))) unsigned int v4u;
typedef __attribute__((ext_vector_type(8))) int v8i;
typedef __attribute__((ext_vector_type(4))) int v4i;

#if __has_builtin(__builtin_amdgcn_tensor_load_to_lds) && \
    __has_builtin(__builtin_amdgcn_s_wait_tensorcnt)
#define HAVE_TDM 1
#else
#define HAVE_TDM 0
#endif

// ---------------------------------------------------------------------------
// agg = x   (folds the GIN self-term x + sum_j x_j into the accumulator init)
// ---------------------------------------------------------------------------
__global__ __launch_bounds__(256)
void gin_init_agg(const float* __restrict__ x, float* __restrict__ agg, int n) {
  int i = blockIdx.x * blockDim.x + threadIdx.x;
  int total = n * (DIM / 4);
  if (i < total) ((f32x4*)agg)[i] = ((const f32x4*)x)[i];
}

// ---------------------------------------------------------------------------
// agg[dst[e]] += x[src[e]]  — one wave32 per edge, float4 per lane,
// 4x global_atomic_add_f32 (destination array is L2-resident: 25.6 MB << 192 MB)
// ---------------------------------------------------------------------------
__global__ __launch_bounds__(256)
void gin_scatter_add(const float* __restrict__ x, const int* __restrict__ src,
                     const int* __restrict__ dst, float* __restrict__ agg, int e) {
  int wid  = (int)(((size_t)blockIdx.x * blockDim.x + threadIdx.x) >> 5);
  int lane = threadIdx.x & 31;
  if (wid >= e) return;
  int s = src[wid];
  int d = dst[wid];
  f32x4 v = ((const f32x4*)(x + (size_t)s * DIM))[lane];
  float* p = agg + (size_t)d * DIM + lane * 4;
  atomicAdd(p + 0, v.x);
  atomicAdd(p + 1, v.y);
  atomicAdd(p + 2, v.z);
  atomicAdd(p + 3, v.w);
}

// ---------------------------------------------------------------------------
// Per-wave strip GEMM: accs[nt] += Strip(16xK=128) @ W[:, nt*16..+16)
// via V_WMMA_F32_16X16X4_F32 (exact fp32; K=4/op). One A fragment feeds 8
// independent WMMA chains per k-step for XDL pipelining.
// ---------------------------------------------------------------------------
#define LDR 132  // padded LDS row stride: 528B = 16B aligned, conflict-free cols

__device__ __forceinline__
void strip_gemm(const float* __restrict__ S,        // wave's 16xDIM strip (LDS)
                const float* __restrict__ W,        // DIMxDIM row-major, global
                const float* __restrict__ bias,     // DIM
                int nl, int kh, v8f accs[8]) {
  #pragma unroll
  for (int nt = 0; nt < 8; ++nt) {
    float bv = bias[nt * 16 + nl];
    #pragma unroll
    for (int r = 0; r < 8; ++r) accs[nt][r] = bv;
  }
  for (int k0 = 0; k0 < DIM; k0 += 4) {
    v2f a;
    a.x = S[nl * LDR + k0 + kh];
    a.y = S[nl * LDR + k0 + kh + 1];
    const float* w0 = W + (size_t)(k0 + kh) * DIM + nl;
    const float* w1 = W + (size_t)(k0 + kh + 1) * DIM + nl;
    #pragma unroll
    for (int nt = 0; nt < 8; ++nt) {
      v2f b;
      b.x = w0[nt * 16];
      b.y = w1[nt * 16];
      accs[nt] = __builtin_amdgcn_wmma_f32_16x16x4_f32(
          false, a, false, b, (short)0, accs[nt], false, false);
    }
  }
}

// ---------------------------------------------------------------------------
// Fused GIN MLP: out = relu( relu(agg @ W1 + b1) @ W2 + b2 )
// 128 threads = 4 waves; each wave owns a private 16-row LDS strip staged by
// ONE Tensor-Data-Mover op (tensor_load_to_lds), completion tracked with the
// wave-private TENSORcnt -> no workgroup barriers anywhere in the kernel.
// ---------------------------------------------------------------------------
__global__ __launch_bounds__(128)
void gin_mlp(const float* __restrict__ agg,
             const float* __restrict__ W1, const float* __restrict__ b1,
             const float* __restrict__ W2, const float* __restrict__ b2,
             float* __restrict__ out, int n) {
  __shared__ float lS[4 * 16 * LDR];          // 33.8 KB

  const int tid  = threadIdx.x;
  const int wave = tid >> 5;
  const int lane = tid & 31;
  const int nl   = lane & 15;                 // N col in tile; M row for A frag
  const int kh   = (lane >> 4) << 1;          // 0 or 2 (K sub-offset)
  const int mh   = (lane >> 4) << 3;          // 0 or 8 (C/D row offset)
  const int rowBase = blockIdx.x * 64 + wave * 16;

  // Warm the far cache with the second-GEMM weights while we stage A.
  __builtin_prefetch(W2 + (size_t)tid * DIM, 0, 1);

  float* S = lS + wave * 16 * LDR;

#if HAVE_TDM
  // ---- TDM: one 2D tile descriptor per wave: 16 rows x 128 f32 ----
  // D# group0: count=1 | lds_addr | global_addr(tile start) | type=2
  const uint32_t lds_off = (uint32_t)(uintptr_t)S;          // addrspace(3) offset
  const uint64_t ga      = (uint64_t)(uintptr_t)(agg + (size_t)rowBase * DIM);
  v4u g0;
  g0.x = 1u;                                                // count=1, user mode
  g0.y = lds_off;                                           // bits[63:32]
  g0.z = (uint32_t)ga;                                      // global_addr lo
  g0.w = (uint32_t)(ga >> 32) | (2u << 30);                 // addr[56:32] | type=2
  // D# group1: data_size=4B; LDS pad: every 128 DWORDs (code 6) add 4 DWORDs
  // (code 3)  =>  row stride 132 == LDR. Tail rows: tensor_dim1 = n-rowBase,
  // OOB rows read back as zero (stores of those rows are masked anyway).
  const uint32_t remRows = (uint32_t)(n - rowBase);
  v8i g1;
  g1[0] = (int)((2u << 16) | (1u << 20) | (6u << 22) | (3u << 25));
  g1[1] = (int)(128u << 16);        // tensor_dim0 = 128 (lo16 in bits[63:48])
  g1[2] = (int)(remRows << 16);     // tensor_dim0 hi=0 | tensor_dim1 lo16
  g1[3] = (int)(128u << 16);        // tensor_dim1 hi=0 | tile_dim0 = 128
  g1[4] = 16;                       // tile_dim1 = 16, tile_dim2 = 0
  g1[5] = 128;                      // tensor_dim0_stride = 128 (lo32)
  g1[6] = 0;                        // stride hi | tensor_dim1_stride lo (unused)
  g1[7] = 0;
  v4i gz4 = {0, 0, 0, 0};           // groups 2/3 unused (2D tensor)
  v8i gz8 = {0, 0, 0, 0, 0, 0, 0, 0};
  __builtin_amdgcn_tensor_load_to_lds(g0, g1, gz4, gz4, gz8, 0);
  __builtin_amdgcn_s_wait_tensorcnt(0);       // wave-private completion
  asm volatile("" ::: "memory");              // order LDS reads after the wait
#else
  // Fallback: plain b128 global->register->LDS staging.
  for (int i = lane; i < 16 * (DIM / 4); i += 32) {
    int r = i >> 5;
    int c = i & 31;
    int gr = rowBase + r;
    if (gr >= n) gr = n - 1;
    f32x4 v = ((const f32x4*)agg)[(size_t)gr * (DIM / 4) + c];
    *(f32x4*)(S + r * LDR + c * 4) = v;
  }
#endif

  v8f accs[8];

  // GEMM1 + ReLU -> overwrite strip in place (all A reads complete first;
  // wave-private strip => DScnt ordering suffices, no barrier).
  strip_gemm(S, W1, b1, nl, kh, accs);
  #pragma unroll
  for (int nt = 0; nt < 8; ++nt)
    #pragma unroll
    for (int r = 0; r < 8; ++r) {
      float v = accs[nt][r];
      S[(r + mh) * LDR + nt * 16 + nl] = v > 0.0f ? v : 0.0f;
    }

  // GEMM2 + ReLU -> global, tail-masked
  strip_gemm(S, W2, b2, nl, kh, accs);
  #pragma unroll
  for (int nt = 0; nt < 8; ++nt)
    #pragma unroll
    for (int r = 0; r < 8; ++r) {
      int gm = rowBase + r + mh;
      if (gm < n) {
        float v = accs[nt][r];
        out[(size_t)gm * DIM + nt * 16 + nl] = v > 0.0f ? v : 0.0f;
      }
    }
}

// ---------------------------------------------------------------------------
extern "C" void kernel_launch(void* const* d_in, const int* in_sizes, int n_in,
                              void* d_out, int out_size, void* d_ws, size_t ws_size,
                              hipStream_t stream) {
  const float* x   = (const float*)d_in[0];
  const int*   ei  = (const int*)d_in[1];
  const float* W1a = (const float*)d_in[2];
  const float* b1a = (const float*)d_in[3];
  const float* W2a = (const float*)d_in[4];
  const float* b2a = (const float*)d_in[5];
  const float* W1b = (const float*)d_in[6];
  const float* b1b = (const float*)d_in[7];
  const float* W2b = (const float*)d_in[8];
  const float* b2b = (const float*)d_in[9];
  float* out = (float*)d_out;

  const int n = in_sizes[0] / DIM;   // 50000
  const int e = in_sizes[1] / 2;     // 800000
  const int* src = ei;
  const int* dst = ei + e;

  float* agg = (float*)d_ws;                          // N*DIM f32
  float* h   = agg + (size_t)n * DIM;                 // N*DIM f32

  const int initBlocks = (n * (DIM / 4) + 255) / 256;
  const int scatBlocks = (e + 7) / 8;                 // 8 waves / 256-thr block
  const int mlpBlocks  = (n + 63) / 64;               // 64 rows / block

  // ---- layer 0 ----
  gin_init_agg  <<<initBlocks, 256, 0, stream>>>(x, agg, n);
  gin_scatter_add<<<scatBlocks, 256, 0, stream>>>(x, src, dst, agg, e);
  gin_mlp       <<<mlpBlocks, 128, 0, stream>>>(agg, W1a, b1a, W2a, b2a, h, n);
  // ---- layer 1 ----
  gin_init_agg  <<<initBlocks, 256, 0, stream>>>(h, agg, n);
  gin_scatter_add<<<scatBlocks, 256, 0, stream>>>(h, src, dst, agg, e);
  gin_mlp       <<<mlpBlocks, 128, 0, stream>>>(agg, W1b, b1b, W2b, b2b, out, n);
}